// Lvar_89000312307759
// MI455X (gfx1250) — compile-verified
//
#include <hip/hip_runtime.h>
#include <math.h>

// Problem constants (match reference)
#define NCLS      5
#define N_SP      1024
#define NUM_SEG   (N_SP * NCLS)     // 5120 segments per image
#define IGNORE_LB 255
#define NB        4                  // batch
#define ND        64                 // channels
#define HW        (512 * 512)        // pixels per image (2^18)

typedef __attribute__((ext_vector_type(2))) float v2f;
typedef __attribute__((ext_vector_type(8))) float v8f;

// ---- workspace layout (4-byte words) -------------------------------------
// [0..3]    int   mx[NB]        per-image max index
// [8]       float loss          scalar accumulator
// [16..]    float cnt  [NB*NUM_SEG]
// then      float sumsq[NB*NUM_SEG]          (sum over pixels of sum_d f^2)
// then      float ssum [NB*NUM_SEG*ND]       (per-channel segment sums)
#define MX_OFF    0
#define LOSS_OFF  8
#define CNT_OFF   16
#define SQ_OFF    (CNT_OFF + NB * NUM_SEG)
#define SSUM_OFF  (SQ_OFF + NB * NUM_SEG)
#define WS_WORDS  (SSUM_OFF + NB * NUM_SEG * ND)   // ~1.35M words = 5.4 MB

// --------------------------------------------------------------------------
__global__ void __launch_bounds__(256) k_zero(float* __restrict__ ws) {
    int i = blockIdx.x * blockDim.x + threadIdx.x;
    int stride = gridDim.x * blockDim.x;
    for (int j = i; j < WS_WORDS; j += stride) ws[j] = 0.0f;  // int 0 == bits of 0.0f
}

// Per-image max of indexes: 256 blocks/image, 1024 px/block, 1 atomic/block.
__global__ void __launch_bounds__(256) k_max(const int* __restrict__ idx,
                                             int* __restrict__ mx) {
    int b     = blockIdx.x >> 8;
    int chunk = blockIdx.x & 255;
    const int* p = idx + (size_t)b * HW + (size_t)chunk * 1024;
    int m = 0;
    for (int j = threadIdx.x; j < 1024; j += 256) m = max(m, p[j]);
    for (int off = 16; off; off >>= 1) m = max(m, __shfl_xor(m, off, 32));
    __shared__ int sm[8];
    if ((threadIdx.x & 31) == 0) sm[threadIdx.x >> 5] = m;
    __syncthreads();
    if (threadIdx.x == 0) {
        int mm = sm[0];
        #pragma unroll
        for (int w = 1; w < 8; ++w) mm = max(mm, sm[w]);
        atomicMax(&mx[b], mm);
    }
}

// Single coalesced pass over features. lane <-> consecutive pixel, loop over d.
// Accumulators (5.4 MB total) are L2-resident -> atomics never hit HBM.
__global__ void __launch_bounds__(256) k_accum(const float* __restrict__ feat,
                                               const int*   __restrict__ lbl,
                                               const int*   __restrict__ idx,
                                               const int*   __restrict__ mx,
                                               float* __restrict__ cnt,
                                               float* __restrict__ sumsq,
                                               float* __restrict__ ssum) {
    int p  = blockIdx.x * blockDim.x + threadIdx.x;   // global pixel 0..NB*HW-1
    int b  = p >> 18;                                 // / HW
    int pp = p & (HW - 1);
    int l  = lbl[p];
    int ix = idx[p];
    int m  = mx[b];
    int seg = (l == IGNORE_LB) ? 0 : (m * l + ix);

    const float* f    = feat + (size_t)b * ND * HW + pp;
    float*       srow = ssum + ((size_t)(b * NUM_SEG + seg)) * ND;

    float sq = 0.0f;
    #pragma unroll 4
    for (int d = 0; d < ND; ++d) {
        float v = f[(size_t)d * HW];   // coalesced across lanes
        sq += v * v;
        atomicAdd(&srow[d], v);
    }
    atomicAdd(&cnt[b * NUM_SEG + seg], 1.0f);
    atomicAdd(&sumsq[b * NUM_SEG + seg], sq);
}

__device__ __forceinline__ float pick8(v8f v, int i) {
    float r = v[0];
    #pragma unroll
    for (int j = 1; j < 8; ++j) r = (i == j) ? v[j] : r;
    return r;
}

// One wave per 16-segment tile: Gram diagonal ||ssum_row||^2 via
// V_WMMA_F32_16X16X4_F32 accumulating S*S^T over K=64 in 16 steps.
// A (16x4, lanes 0-15: K={k,k+1}, lanes 16-31: K={k+2,k+3}) and
// B (4x16) have register-identical layouts for S*S^T -> pass `a` twice.
__global__ void __launch_bounds__(32) k_var(const float* __restrict__ cnt,
                                            const float* __restrict__ sumsq,
                                            const float* __restrict__ ssum,
                                            float* __restrict__ loss) {
    int tile = blockIdx.x;                 // 0 .. NB*NUM_SEG/16 - 1
    int b    = tile / (NUM_SEG / 16);
    int t    = tile % (NUM_SEG / 16);
    int lane = threadIdx.x;
    int half = lane >> 4;
    int mrow = lane & 15;
    int seg0 = t * 16;

    const float* Sb = ssum + ((size_t)(b * NUM_SEG + seg0 + mrow)) * ND;

    v8f acc = {};
    #pragma unroll
    for (int k = 0; k < ND; k += 4) {
        v2f a;
        a.x = Sb[k + 2 * half];
        a.y = Sb[k + 2 * half + 1];
        acc = __builtin_amdgcn_wmma_f32_16x16x4_f32(
            /*neg_a=*/false, a, /*neg_b=*/false, a,
            /*c_mod=*/(short)0, acc, /*reuse_a=*/false, /*reuse_b=*/false);
    }

    // Diagonal D[m][m]: m in 0..7 -> VGPR m @ lane m ; m in 8..15 -> VGPR m-8 @ lane m+16.
    float term = 0.0f;
    int  m   = -1;
    float dg = 0.0f;
    if (lane < 8)        { m = lane;      dg = pick8(acc, lane);      }
    else if (lane >= 24) { m = lane - 16; dg = pick8(acc, lane - 24); }
    if (m >= 0) {
        int sid = seg0 + m;
        float c  = cnt[b * NUM_SEG + sid];
        float sq = sumsq[b * NUM_SEG + sid];
        if (c > 1.0f && sid != 0)
            term = (sq - dg / c) / c * (5.0f / 64.0f);
    }
    for (int off = 16; off; off >>= 1) term += __shfl_xor(term, off, 32);
    if (lane == 0) atomicAdd(loss, term);
}

__global__ void k_sqrt(const float* __restrict__ loss, float* __restrict__ out) {
    if (threadIdx.x == 0 && blockIdx.x == 0) out[0] = sqrtf(loss[0]);
}

// --------------------------------------------------------------------------
extern "C" void kernel_launch(void* const* d_in, const int* in_sizes, int n_in,
                              void* d_out, int out_size, void* d_ws, size_t ws_size,
                              hipStream_t stream) {
    const float* feat = (const float*)d_in[0];
    const int*   lbl  = (const int*)d_in[1];
    const int*   idx  = (const int*)d_in[2];
    float* ws   = (float*)d_ws;
    float* out  = (float*)d_out;

    int*   mx    = (int*)(ws + MX_OFF);
    float* loss  = ws + LOSS_OFF;
    float* cnt   = ws + CNT_OFF;
    float* sumsq = ws + SQ_OFF;
    float* ssum  = ws + SSUM_OFF;

    k_zero <<<512, 256, 0, stream>>>(ws);
    k_max  <<<NB * 256, 256, 0, stream>>>(idx, mx);
    k_accum<<<(NB * HW) / 256, 256, 0, stream>>>(feat, lbl, idx, mx, cnt, sumsq, ssum);
    k_var  <<<(NB * NUM_SEG) / 16, 32, 0, stream>>>(cnt, sumsq, ssum, loss);
    k_sqrt <<<1, 1, 0, stream>>>(loss, out);
}